// SimpleModulRNN_5514738008112
// MI455X (gfx1250) — compile-verified
//
#include <hip/hip_runtime.h>
#include <hip/hip_bf16.h>
#include <math.h>

#define BS      256
#define HS      512
#define IN_DIM  64
#define OUT_DIM 4

typedef __attribute__((ext_vector_type(2))) float v2f;
typedef __attribute__((ext_vector_type(4))) float v4f;
typedef __attribute__((ext_vector_type(8))) float v8f;

__device__ __forceinline__ float clip1(float x) {
    return fminf(1.0f, fmaxf(-1.0f, x));
}

// ---------------------------------------------------------------------------
// Kernel A: pre[b,i] = bi[i] + sum_k inputs[b,k]*Wi[i,k] + sum_j hidden[b,j]*w[i,j]
// One 16x16 output tile per wave32 using V_WMMA_F32_16X16X4_F32 (exact f32).
// A-matrix 16x4 layout: lane 0-15 -> M, VGPR0/1 -> K=0,1 ; lane 16-31 -> K=2,3.
// B-matrix 4x16 layout mirrors it with N across lanes. C/D: VGPR r, lane l ->
// M = r + 8*(l>>4), N = l&15.
// Grid: 64 blocks x 256 threads = 512 waves = (256/16)*(512/16) tiles.
// ---------------------------------------------------------------------------
__global__ __launch_bounds__(256) void gemm_pre_kernel(
    const float* __restrict__ inputs,  // [BS, IN_DIM]
    const float* __restrict__ hidden,  // [BS, HS]
    const float* __restrict__ Wi,      // [HS, IN_DIM]
    const float* __restrict__ bi,      // [HS]
    const float* __restrict__ w,       // [HS, HS]
    float* __restrict__ pre)           // [BS, HS]
{
    const int wave = threadIdx.x >> 5;
    const int lane = threadIdx.x & 31;
    const int tile = blockIdx.x * 8 + wave;     // 0..511
    const int tileM = tile >> 5;                // 16 tiles along batch
    const int tileN = tile & 31;                // 32 tiles along HS
    const int m0 = tileM * 16;
    const int n0 = tileN * 16;
    const int r  = lane & 15;                   // M row (A) / N row (B)
    const int kh = lane >> 4;                   // K-half select (0/1)

    v8f acc = {};

    // inputs @ Wi.T contribution (K = 64)
    {
        const float* arow = inputs + (m0 + r) * IN_DIM + 2 * kh;
        const float* brow = Wi     + (n0 + r) * IN_DIM + 2 * kh;
        for (int k0 = 0; k0 < IN_DIM; k0 += 4) {
            v2f a = *(const v2f*)(arow + k0);
            v2f b = *(const v2f*)(brow + k0);
            acc = __builtin_amdgcn_wmma_f32_16x16x4_f32(
                false, a, false, b, (short)0, acc, false, false);
        }
    }
    // hidden @ w.T contribution (K = 512)
    {
        const float* arow = hidden + (m0 + r) * HS + 2 * kh;
        const float* brow = w      + (n0 + r) * HS + 2 * kh;
        for (int k0 = 0; k0 < HS; k0 += 4) {
            v2f a = *(const v2f*)(arow + k0);
            v2f b = *(const v2f*)(brow + k0);
            acc = __builtin_amdgcn_wmma_f32_16x16x4_f32(
                false, a, false, b, (short)0, acc, false, false);
        }
    }

    const float bias = bi[n0 + r];   // column index n == lane&15 == r
#pragma unroll
    for (int rr = 0; rr < 8; ++rr) {
        const int m = m0 + rr + 8 * kh;
        pre[m * HS + (n0 + r)] = acc[rr] + bias;
    }
}

// ---------------------------------------------------------------------------
// Kernel B: fused per-batch pipeline. One workgroup (512 threads = 16 waves)
// per batch element b:
//   phase 1: hactiv[b,i] = tanh(pre[b,i] + sum_j alpha[i,j]*hebb[b,i,j]*hidden[b,j])
//   phase 2: activout / valueout / DAout head reductions
//   phase 3: hebb_new[b,i,j] = clip(hebb[b,i,j] + DA*hactiv[i]*hidden[j], -1, 1)
// hebb[b] (1 MB) is read in phase 1 (regular-temporal, wants L2 residency) and
// re-read in phase 3 with a non-temporal/last-use hint; hebb_new stores are
// non-temporal so the 256 MB write stream does not evict the phase-1 data.
// ---------------------------------------------------------------------------
__global__ __launch_bounds__(512) void fused_batch_kernel(
    const float* __restrict__ hidden,   // [BS, HS]
    const float* __restrict__ hebb,     // [BS, HS, HS]
    const float* __restrict__ alpha,    // [HS, HS]
    const float* __restrict__ pre,      // [BS, HS]
    const float* __restrict__ Wo,       // [OUT_DIM, HS]
    const float* __restrict__ bo,       // [OUT_DIM]
    const float* __restrict__ Wv,       // [1, HS]
    const float* __restrict__ bv,       // [1]
    const float* __restrict__ Wda,      // [1, HS]
    const float* __restrict__ bda,      // [1]
    float* __restrict__ activout,       // [BS, OUT_DIM]
    float* __restrict__ valueout,       // [BS]
    float* __restrict__ daout,          // [BS]
    float* __restrict__ hactiv_out,     // [BS, HS]
    float* __restrict__ hebb_new)       // [BS, HS, HS]
{
    const int b    = blockIdx.x;
    const int t    = threadIdx.x;       // 0..511
    const int wave = t >> 5;            // 0..15
    const int lane = t & 31;

    __shared__ float sh_h[HS];          // hactiv row for this batch
    __shared__ float sh_hid[HS];        // hidden row cached
    __shared__ float red[16];
    __shared__ float sh_da;

    sh_hid[t] = hidden[b * HS + t];
    __syncthreads();

    // ---- phase 1: recurrent plastic term + tanh -------------------------
    for (int i = wave; i < HS; i += 16) {
        const float* hrow = hebb  + ((size_t)b * HS + i) * HS;
        const float* arow = alpha + (size_t)i * HS;
        float s = 0.0f;
#pragma unroll
        for (int c = 0; c < 4; ++c) {
            const int j = c * 128 + lane * 4;
            v4f hv = *(const v4f*)(hrow + j);
            v4f av = *(const v4f*)(arow + j);
            s += av.x * hv.x * sh_hid[j + 0];
            s += av.y * hv.y * sh_hid[j + 1];
            s += av.z * hv.z * sh_hid[j + 2];
            s += av.w * hv.w * sh_hid[j + 3];
        }
#pragma unroll
        for (int off = 16; off; off >>= 1) s += __shfl_xor(s, off, 32);
        if (lane == 0) sh_h[i] = tanhf(pre[b * HS + i] + s);
    }
    __syncthreads();

    hactiv_out[b * HS + t] = sh_h[t];

    // ---- phase 2: output heads ------------------------------------------
    const float h0 = sh_h[t];
    const float wv6[6] = { Wo[0 * HS + t], Wo[1 * HS + t], Wo[2 * HS + t],
                           Wo[3 * HS + t], Wv[t], Wda[t] };
    for (int k = 0; k < 6; ++k) {
        float s = h0 * wv6[k];
#pragma unroll
        for (int off = 16; off; off >>= 1) s += __shfl_xor(s, off, 32);
        if (lane == 0) red[wave] = s;
        __syncthreads();
        if (t == 0) {
            float tot = 0.0f;
#pragma unroll
            for (int q = 0; q < 16; ++q) tot += red[q];
            if (k < 4) {
                activout[b * OUT_DIM + k] = tot + bo[k];
            } else if (k == 4) {
                valueout[b] = tot + bv[0];
            } else {
                const float d = tanhf(tot + bda[0]);
                daout[b] = d;
                sh_da   = d;
            }
        }
        __syncthreads();
    }

    // ---- phase 3: plastic weight update (hebb[b] should be L2-hot) ------
    const float da = sh_da;
    for (int i = wave; i < HS; i += 16) {
        const float scale = da * sh_h[i];
        const float* hrow = hebb     + ((size_t)b * HS + i) * HS;
        float*       orow = hebb_new + ((size_t)b * HS + i) * HS;
#pragma unroll
        for (int c = 0; c < 4; ++c) {
            const int j = c * 128 + lane * 4;
            // last use of this hebb line -> non-temporal load hint
            v4f hv = __builtin_nontemporal_load((const v4f*)(hrow + j));
            v4f rr;
            rr.x = clip1(hv.x + scale * sh_hid[j + 0]);
            rr.y = clip1(hv.y + scale * sh_hid[j + 1]);
            rr.z = clip1(hv.z + scale * sh_hid[j + 2]);
            rr.w = clip1(hv.w + scale * sh_hid[j + 3]);
            // output is never re-read -> non-temporal store hint
            __builtin_nontemporal_store(rr, (v4f*)(orow + j));
        }
    }
}

// ---------------------------------------------------------------------------
extern "C" void kernel_launch(void* const* d_in, const int* in_sizes, int n_in,
                              void* d_out, int out_size, void* d_ws, size_t ws_size,
                              hipStream_t stream)
{
    const float* inputs = (const float*)d_in[0];   // [256,64]
    const float* hidden = (const float*)d_in[1];   // [256,512]
    const float* hebb   = (const float*)d_in[2];   // [256,512,512]
    const float* Wi     = (const float*)d_in[3];   // [512,64]
    const float* bi     = (const float*)d_in[4];   // [512]
    const float* w      = (const float*)d_in[5];   // [512,512]
    const float* alpha  = (const float*)d_in[6];   // [512,512]
    const float* Wo     = (const float*)d_in[7];   // [4,512]
    const float* bo     = (const float*)d_in[8];   // [4]
    const float* Wv     = (const float*)d_in[9];   // [1,512]
    const float* bv     = (const float*)d_in[10];  // [1]
    const float* Wda    = (const float*)d_in[11];  // [1,512]
    const float* bda    = (const float*)d_in[12];  // [1]

    float* out       = (float*)d_out;
    float* activout  = out;                              // 256*4
    float* valueout  = out + BS * OUT_DIM;               // 256
    float* daout     = valueout + BS;                    // 256
    float* hactiv    = daout + BS;                       // 256*512
    float* hebb_new  = hactiv + BS * HS;                 // 256*512*512

    float* pre = (float*)d_ws;                           // [256,512] = 512 KB

    // A: WMMA GEMMs for the batch-shared linear parts
    gemm_pre_kernel<<<64, 256, 0, stream>>>(inputs, hidden, Wi, bi, w, pre);

    // B: fused per-batch plasticity pipeline (one WG per batch element)
    fused_batch_kernel<<<BS, 512, 0, stream>>>(
        hidden, hebb, alpha, pre,
        Wo, bo, Wv, bv, Wda, bda,
        activout, valueout, daout, hactiv, hebb_new);
}